// EquationLayer_11527692222816
// MI455X (gfx1250) — compile-verified
//
#include <hip/hip_runtime.h>

#define F 32
#define P 496
#define NCOL 3696

// workspace / LDS constant-block layout (dword offsets)
#define LINK_OFF 0        // linear knots  [3][F][R] -> 128 + 512 + 2048 = 2688
#define CUBK_OFF 2688     // cubic knots   2688
#define M_OFF    5376     // cubic M       2688
#define FW_OFF   8064     // masked lm[32], cm[32], rm[32]
#define PW_OFF   8160     // masked lpm[496], cpm[496], rpm[496]
#define LUT_OFF  9648     // pair LUT (i<<5|j) as raw bits, 496 (16B aligned, 496%4==0)
#define TOTAL_DW 10144    // 40576 bytes, divisible by 4 dwords -> b128 copies

#define SV_STRIDE 12      // dwords per feature slot-vector (8 used + 4 pad), 48B

__device__ __forceinline__ float maskf(float w) {
    float a = fabsf(w);
    return (a > 1e-07f) ? a : 0.0f;
}

__device__ __forceinline__ unsigned lds_u32(void* p) {
    return (unsigned)(unsigned long long)(__attribute__((address_space(3))) char*)p;
}

// ---------------------------------------------------------------------------
// Precompute: pack knots + masked weights + pair LUT + cubic-spline M arrays
// into a contiguous workspace block. One block, runs once per launch.
// ---------------------------------------------------------------------------
__global__ __launch_bounds__(256) void eq_precompute(
    const float* __restrict__ linear_fw, const float* __restrict__ cubic_fw,
    const float* __restrict__ raw_fw,    const float* __restrict__ linear_pw,
    const float* __restrict__ cubic_pw,  const float* __restrict__ raw_pw,
    const float* __restrict__ lk0, const float* __restrict__ ck0,
    const float* __restrict__ lk1, const float* __restrict__ ck1,
    const float* __restrict__ lk2, const float* __restrict__ ck2,
    float* __restrict__ ws)
{
    __shared__ float cp[64];
    __shared__ float dp[96 * 64];
    const int tid = threadIdx.x;

    const float* lks[3] = { lk0, lk1, lk2 };
    const float* cks[3] = { ck0, ck1, ck2 };
    const int sz[3]  = { 128, 512, 2048 };
    const int off[3] = { 0, 128, 640 };

    // knots -> contiguous block
    for (int r = 0; r < 3; ++r)
        for (int k = tid; k < sz[r]; k += 256) {
            ws[LINK_OFF + off[r] + k] = lks[r][k];
            ws[CUBK_OFF + off[r] + k] = cks[r][k];
        }

    // masked per-feature weights
    if (tid < 32) {
        ws[FW_OFF + tid]      = maskf(linear_fw[tid]);
        ws[FW_OFF + 32 + tid] = maskf(cubic_fw[tid]);
        ws[FW_OFF + 64 + tid] = maskf(raw_fw[tid]);
    }
    // masked pair weights + pair index LUT
    for (int p = tid; p < P; p += 256) {
        ws[PW_OFF + p]       = maskf(linear_pw[p]);
        ws[PW_OFF + 496 + p] = maskf(cubic_pw[p]);
        ws[PW_OFF + 992 + p] = maskf(raw_pw[p]);
        int i = 0;
        while ((i + 1) * 32 - ((i + 1) * (i + 2)) / 2 <= p) ++i;  // offset(i+1)<=p
        int baseo = i * 32 - (i * (i + 1)) / 2;
        int j = i + 1 + (p - baseo);
        ws[LUT_OFF + p] = __uint_as_float((unsigned)((i << 5) | j));
    }

    // data-independent Thomas coefficients for (1,4,1) system
    if (tid == 0) {
        cp[0] = 0.25f;
        for (int i = 1; i < 62; ++i) cp[i] = 1.0f / (4.0f - cp[i - 1]);
    }
    __syncthreads();

    // natural cubic spline: solve A*M_int = rhs per (resolution, feature)
    if (tid < 96) {
        const int r = tid >> 5, f = tid & 31;
        const int R = (r == 0) ? 4 : ((r == 1) ? 16 : 64);
        const int n = R - 2;
        const float* y = cks[r] + f * R;
        const float c6 = (float)(6 * (R - 1) * (R - 1));   // 6/h^2, exact integer
        float* d = &dp[tid * 64];
        d[0] = ((y[2] - 2.0f * y[1]) + y[0]) * c6 * 0.25f;
        for (int i = 1; i < n; ++i) {
            float rhs = ((y[i + 2] - 2.0f * y[i + 1]) + y[i]) * c6;
            d[i] = (rhs - d[i - 1]) * cp[i];
        }
        float* Mrow = &ws[M_OFF + off[r] + f * R];
        Mrow[0] = 0.0f; Mrow[R - 1] = 0.0f;
        float xn = d[n - 1];
        Mrow[n] = xn;                       // index 1+(n-1)
        for (int i = n - 2; i >= 0; --i) {
            xn = d[i] - cp[i] * xn;
            Mrow[1 + i] = xn;
        }
    }
}

// ---------------------------------------------------------------------------
// Main streaming kernel: 8 waves/block, wave-per-row, lane-per-feature.
// Const tables staged into LDS via CDNA5 async global->LDS DMA.
// Pair stage uses feature-major LDS tiles (b128 gathers) + b128 stores.
// ---------------------------------------------------------------------------
__global__ __launch_bounds__(256) void eq_main(const float* __restrict__ in,
                                               const float* __restrict__ ws,
                                               float* __restrict__ out,
                                               int nrows)
{
    __shared__ __align__(16) float sc[TOTAL_DW];
    __shared__ __align__(16) float sv[8][F * SV_STRIDE];
    const int tid = threadIdx.x;

    // ---- async stage: ws -> LDS, 16B per lane per issue ----
    for (int k = tid; k < TOTAL_DW / 4; k += 256) {
        unsigned laddr = lds_u32((void*)&sc[k * 4]);
        unsigned long long gaddr = (unsigned long long)(const void*)(ws + k * 4);
        asm volatile("global_load_async_to_lds_b128 %0, %1, off"
                     :: "v"(laddr), "v"(gaddr) : "memory");
    }
    asm volatile("s_wait_asynccnt 0" ::: "memory");
    __syncthreads();

    const int w = tid >> 5, lane = tid & 31;
    const float lm  = sc[FW_OFF + lane];
    const float cmk = sc[FW_OFF + 32 + lane];
    const float rmk = sc[FW_OFF + 64 + lane];
    float* svg = &sv[w][0];

    for (int it = 0; it < 4; ++it) {
        const int row = blockIdx.x * 32 + it * 8 + w;
        if (row < nrows) {
            const long long base = (long long)row * NCOL;
            const float x = in[row * F + lane];
            const float xc = fminf(fmaxf(x, 0.0f), 1.0f);

            out[base + lane] = x * rmk;   // raw unary (masked)

            float vlin[3], vcub[3];
#pragma unroll
            for (int j = 0; j < 3; ++j) {
                const int R = (j == 0) ? 4 : ((j == 1) ? 16 : 64);
                const int O = (j == 0) ? 0 : ((j == 1) ? 128 : 640);
                // -------- linear spline --------
                {
                    float t = xc * (float)(R - 1);
                    int idx = (int)floorf(t);
                    idx = idx < 0 ? 0 : (idx > R - 2 ? R - 2 : idx);
                    float frac = t - (float)idx;
                    const float* kr = &sc[LINK_OFF + O + lane * R];
                    float y0 = kr[idx], y1 = kr[idx + 1];
                    float v = (y0 * (1.0f - frac) + y1 * frac) * lm;
                    out[base + 32 + j * 32 + lane] = v;
                    vlin[j] = v;
                }
                // -------- natural cubic spline --------
                {
                    const float hf = 1.0f / (float)(R - 1);
                    int idx = (int)floorf(xc / hf);
                    idx = idx < 0 ? 0 : (idx > R - 2 ? R - 2 : idx);
                    float u = xc - (float)idx * hf;
                    const float* kr = &sc[CUBK_OFF + O + lane * R];
                    const float* mr = &sc[M_OFF + O + lane * R];
                    float y0 = kr[idx], y1 = kr[idx + 1];
                    float m0 = mr[idx], m1 = mr[idx + 1];
                    float hu = hf - u;
                    float v = (m0 * hu * hu * hu + m1 * u * u * u) / (6.0f * hf)
                            + (y0 / hf - m0 * hf / 6.0f) * hu
                            + (y1 / hf - m1 * hf / 6.0f) * u;
                    v *= cmk;
                    out[base + 128 + j * 32 + lane] = v;
                    vcub[j] = v;
                }
            }

            // feature-major LDS tile: slot vector {x, lin0..2, cub0..2, pad}
            *(float4*)&svg[lane * SV_STRIDE]     = make_float4(x, vlin[0], vlin[1], vlin[2]);
            *(float4*)&svg[lane * SV_STRIDE + 4] = make_float4(vcub[0], vcub[1], vcub[2], 0.0f);

            // -------- pairwise products: 4 consecutive p per lane, b128 I/O ----
            for (int it2 = 0; it2 < 4; ++it2) {
                const int p0 = it2 * 128 + lane * 4;
                if (p0 < P) {
                    float4 lut4 = *(const float4*)&sc[LUT_OFF + p0];
                    float4 wl4  = *(const float4*)&sc[PW_OFF + p0];
                    float4 wc4  = *(const float4*)&sc[PW_OFF + 496 + p0];
                    float4 wr4  = *(const float4*)&sc[PW_OFF + 992 + p0];
                    const float lutc[4] = { lut4.x, lut4.y, lut4.z, lut4.w };
                    const float wlc[4]  = { wl4.x,  wl4.y,  wl4.z,  wl4.w };
                    const float wcc[4]  = { wc4.x,  wc4.y,  wc4.z,  wc4.w };
                    const float wrc[4]  = { wr4.x,  wr4.y,  wr4.z,  wr4.w };
                    float rq[4], l0q[4], l1q[4], l2q[4], c0q[4], c1q[4], c2q[4];
#pragma unroll
                    for (int q = 0; q < 4; ++q) {
                        unsigned ij = __float_as_uint(lutc[q]);
                        int i  = (int)(ij >> 5);
                        int jx = (int)(ij & 31u);
                        float4 A0 = *(const float4*)&svg[i * SV_STRIDE];
                        float4 A1 = *(const float4*)&svg[i * SV_STRIDE + 4];
                        float4 B0 = *(const float4*)&svg[jx * SV_STRIDE];
                        float4 B1 = *(const float4*)&svg[jx * SV_STRIDE + 4];
                        rq[q]  = A0.x * B0.x * wrc[q];
                        l0q[q] = A0.y * B0.y * wlc[q];
                        l1q[q] = A0.z * B0.z * wlc[q];
                        l2q[q] = A0.w * B0.w * wlc[q];
                        c0q[q] = A1.x * B1.x * wcc[q];
                        c1q[q] = A1.y * B1.y * wcc[q];
                        c2q[q] = A1.z * B1.z * wcc[q];
                    }
                    *(float4*)&out[base + 224  + p0] = make_float4(rq[0],  rq[1],  rq[2],  rq[3]);
                    *(float4*)&out[base + 720  + p0] = make_float4(l0q[0], l0q[1], l0q[2], l0q[3]);
                    *(float4*)&out[base + 1216 + p0] = make_float4(l1q[0], l1q[1], l1q[2], l1q[3]);
                    *(float4*)&out[base + 1712 + p0] = make_float4(l2q[0], l2q[1], l2q[2], l2q[3]);
                    *(float4*)&out[base + 2208 + p0] = make_float4(c0q[0], c0q[1], c0q[2], c0q[3]);
                    *(float4*)&out[base + 2704 + p0] = make_float4(c1q[0], c1q[1], c1q[2], c1q[3]);
                    *(float4*)&out[base + 3200 + p0] = make_float4(c2q[0], c2q[1], c2q[2], c2q[3]);
                }
            }
        }
    }
}

extern "C" void kernel_launch(void* const* d_in, const int* in_sizes, int n_in,
                              void* d_out, int out_size, void* d_ws, size_t ws_size,
                              hipStream_t stream) {
    const float* inputs    = (const float*)d_in[0];
    const float* linear_fw = (const float*)d_in[1];
    const float* cubic_fw  = (const float*)d_in[2];
    const float* raw_fw    = (const float*)d_in[3];
    const float* linear_pw = (const float*)d_in[4];
    const float* cubic_pw  = (const float*)d_in[5];
    const float* raw_pw    = (const float*)d_in[6];
    // setup_inputs() insertion order interleaves lin/cub per resolution
    const float* lk0 = (const float*)d_in[7];
    const float* ck0 = (const float*)d_in[8];
    const float* lk1 = (const float*)d_in[9];
    const float* ck1 = (const float*)d_in[10];
    const float* lk2 = (const float*)d_in[11];
    const float* ck2 = (const float*)d_in[12];

    float* ws  = (float*)d_ws;
    float* out = (float*)d_out;
    const int nrows = in_sizes[0] / F;   // 32768

    eq_precompute<<<1, 256, 0, stream>>>(linear_fw, cubic_fw, raw_fw,
                                         linear_pw, cubic_pw, raw_pw,
                                         lk0, ck0, lk1, ck1, lk2, ck2, ws);

    const int grid = (nrows + 31) / 32;  // 32 rows per block (8 waves x 4 iters)
    eq_main<<<grid, 256, 0, stream>>>(inputs, ws, out, nrows);
}